// Schema_Relation_Network_29437705847372
// MI455X (gfx1250) — compile-verified
//
#include <hip/hip_runtime.h>
#include <climits>

#define N_DST 20000
#define N_SRC 40000
#define N_EDGE 320000
#define RAW 256
#define HID 64
#define NH 4
#define HD 64
#define FDIM 256   // NH*HD

typedef __attribute__((ext_vector_type(2))) float v2f;
typedef __attribute__((ext_vector_type(8))) float v8f;

// ---------------------------------------------------------------------------
// Wave-level 16x16 f32 WMMA tile: C += A(16xK) * B(Kx16), A row-major (lda),
// B row-major (ldb). Layout per CDNA5 ISA 7.12.2:
//   A 16x4: lanes 0-15 rows M=0..15 hold K=0 (v0), K=1 (v1);
//           lanes 16-31 hold K=2 (v0), K=3 (v1).
//   B 4x16: lane n holds col n; v0/v1 = rows K=0/1 (lanes<16), K=2/3 (lanes>=16)
//   C/D 16x16: VGPR j = row (j + 8*half), col = lane&15.
// ---------------------------------------------------------------------------
__device__ __forceinline__ v8f wmma16x16_f32(const float* A, int lda,
                                             const float* B, int ldb,
                                             int K, v8f c) {
  const int lane = threadIdx.x & 31;
  const int half = lane >> 4;
  const int r    = lane & 15;
  const float* ap = A + r * lda + 2 * half;
  const float* bp = B + (2 * half) * ldb + r;
  for (int k = 0; k < K; k += 4) {
    v2f a, b;
    a[0] = ap[k];
    a[1] = ap[k + 1];
    b[0] = bp[k * ldb];
    b[1] = bp[(k + 1) * ldb];
    c = __builtin_amdgcn_wmma_f32_16x16x4_f32(false, a, false, b,
                                              (short)0, c, false, false);
  }
  return c;
}

// ---------------------------------------------------------------------------
// Node transform (per relation side):
//   h  = X[M,256] @ W1[256,64] + b1            (f32 WMMA)
//   F  = h @ W2[64,256]                        (f32 WMMA)
//   l  = per-head dot:  l[m,h] = sum_d F[m,h*64+d] * avec[h,d]
// F optionally stored (src side); l always stored.
// Block = 128 threads (4 waves), 16 rows per block.
// ---------------------------------------------------------------------------
__global__ void node_transform_kernel(const float* __restrict__ X,
                                      const float* __restrict__ W1,
                                      const float* __restrict__ b1,
                                      const float* __restrict__ W2,
                                      const float* __restrict__ avec,
                                      float* __restrict__ Fout,   // [M,256] or null
                                      float* __restrict__ lout,   // [M,4]
                                      int M) {
  __shared__ float sX[16 * RAW];   // 16 KB
  __shared__ float sH[16 * HID];   //  4 KB
  __shared__ float sF[16 * FDIM];  // 16 KB
  const int tid  = threadIdx.x;
  const int wave = tid >> 5;
  const int lane = tid & 31;
  const int half = lane >> 4;
  const int cr   = lane & 15;
  const int row0 = blockIdx.x * 16;

  // stage X tile (zero-padded) -- coalesced
  for (int i = tid; i < 16 * RAW; i += 128) {
    int r = i >> 8, cc = i & 255;
    int g = row0 + r;
    sX[i] = (g < M) ? X[(size_t)g * RAW + cc] : 0.0f;
  }
  __syncthreads();

  // phase 1: h tile (wave w -> cols [w*16, w*16+16))
  {
    v8f c = {};
    c = wmma16x16_f32(sX, RAW, W1 + wave * 16, HID, RAW, c);
    float bias = b1[wave * 16 + cr];
#pragma unroll
    for (int j = 0; j < 8; j++) {
      int m = j + 8 * half;
      sH[m * HID + wave * 16 + cr] = c[j] + bias;
    }
  }
  __syncthreads();

  // phase 2: F tile (16 col tiles, 4 per wave)
  for (int t = wave; t < 16; t += 4) {
    v8f c = {};
    c = wmma16x16_f32(sH, HID, W2 + t * 16, FDIM, HID, c);
#pragma unroll
    for (int j = 0; j < 8; j++) {
      int m = j + 8 * half;
      sF[m * FDIM + t * 16 + cr] = c[j];
    }
  }
  __syncthreads();

  if (Fout) {
    for (int i = tid; i < 16 * FDIM; i += 128) {
      int r = i >> 8, cc = i & 255;
      int g = row0 + r;
      if (g < M) Fout[(size_t)g * FDIM + cc] = sF[i];
    }
  }

  // phase 3: per-head attention logits
  if (tid < 64) {
    int r = tid >> 2, h = tid & 3;
    int g = row0 + r;
    if (g < M) {
      float acc = 0.0f;
      for (int d = 0; d < HD; d++)
        acc += sF[r * FDIM + h * HD + d] * avec[h * HD + d];
      lout[(size_t)g * NH + h] = acc;
    }
  }
}

// ---------------------------------------------------------------------------
// Monotone float<->ordered-int mapping for atomic max on f32.
// ---------------------------------------------------------------------------
__device__ __forceinline__ int order_f32(float f) {
  int i = __float_as_int(f);
  return (i >= 0) ? i : (i ^ 0x7fffffff);
}
__device__ __forceinline__ float unorder_f32(int i) {
  return __int_as_float((i >= 0) ? i : (i ^ 0x7fffffff));
}

__global__ void init_relation_kernel(int* __restrict__ mbuf,
                                     float* __restrict__ sbuf,
                                     float* __restrict__ outacc) {
  int i = blockIdx.x * blockDim.x + threadIdx.x;
  if (i < N_DST * FDIM) outacc[i] = 0.0f;
  if (i < N_DST * NH) { mbuf[i] = INT_MIN; sbuf[i] = 0.0f; }
}

// pass 1: e = leaky_relu(el[src]+er[dst]); store e; atomic max per (dst,head)
__global__ void edge_logits_max_kernel(const int* __restrict__ si,
                                       const int* __restrict__ di,
                                       const float* __restrict__ el,
                                       const float* __restrict__ er,
                                       float* __restrict__ ebuf,
                                       int* __restrict__ mbuf) {
  int e = blockIdx.x * blockDim.x + threadIdx.x;
  if (e >= N_EDGE) return;
  int s = si[e], d = di[e];
  float4 l = ((const float4*)el)[s];
  float4 r = ((const float4*)er)[d];
  float v[4] = {l.x + r.x, l.y + r.y, l.z + r.z, l.w + r.w};
  float4 o;
#pragma unroll
  for (int h = 0; h < 4; h++) {
    float x = v[h];
    x = (x > 0.0f) ? x : 0.2f * x;
    v[h] = x;
    atomicMax(&mbuf[d * NH + h], order_f32(x));
  }
  o.x = v[0]; o.y = v[1]; o.z = v[2]; o.w = v[3];
  ((float4*)ebuf)[e] = o;
}

// pass 2: p = exp(e - m[dst]); store p; atomic add per (dst,head)
__global__ void edge_softmax_sum_kernel(const int* __restrict__ di,
                                        float* __restrict__ ebuf,
                                        const int* __restrict__ mbuf,
                                        float* __restrict__ sbuf) {
  int e = blockIdx.x * blockDim.x + threadIdx.x;
  if (e >= N_EDGE) return;
  int d = di[e];
  float4 ev = ((const float4*)ebuf)[e];
  float v[4] = {ev.x, ev.y, ev.z, ev.w};
#pragma unroll
  for (int h = 0; h < 4; h++) {
    float mf = unorder_f32(mbuf[d * NH + h]);
    if (!isfinite(mf)) mf = 0.0f;
    float p = __expf(v[h] - mf);
    v[h] = p;
    atomicAdd(&sbuf[d * NH + h], p);
  }
  float4 o; o.x = v[0]; o.y = v[1]; o.z = v[2]; o.w = v[3];
  ((float4*)ebuf)[e] = o;
}

// pass 3: one wave per edge; lane owns 8 of 256 cols (head = lane/8)
__global__ void edge_aggregate_kernel(const int* __restrict__ si,
                                      const int* __restrict__ di,
                                      const float* __restrict__ ebuf,
                                      const float* __restrict__ sbuf,
                                      const float* __restrict__ F,
                                      float* __restrict__ outacc) {
  int w = (blockIdx.x * blockDim.x + threadIdx.x) >> 5;
  int lane = threadIdx.x & 31;
  if (w >= N_EDGE) return;
  int s = si[w], d = di[w];
  int h = lane >> 3;
  float p  = ebuf[(size_t)w * NH + h];
  float sv = sbuf[(size_t)d * NH + h];
  float a = p / fmaxf(sv, 1e-9f);
  int c0 = lane * 8;
  const float4* fs = (const float4*)(F + (size_t)s * FDIM + c0);
  float4 f0 = fs[0], f1 = fs[1];
  float* o = outacc + (size_t)d * FDIM + c0;
  atomicAdd(o + 0, a * f0.x); atomicAdd(o + 1, a * f0.y);
  atomicAdd(o + 2, a * f0.z); atomicAdd(o + 3, a * f0.w);
  atomicAdd(o + 4, a * f1.x); atomicAdd(o + 5, a * f1.y);
  atomicAdd(o + 6, a * f1.z); atomicAdd(o + 7, a * f1.w);
}

// z[n, rIdx, :] = elu(outacc[n,:] + b)
__global__ void finalize_z_kernel(const float* __restrict__ outacc,
                                  const float* __restrict__ b,
                                  float* __restrict__ z, int rIdx) {
  int i = blockIdx.x * blockDim.x + threadIdx.x;
  if (i >= N_DST * FDIM) return;
  int n = i >> 8, cc = i & 255;
  float x = outacc[i] + b[cc];
  z[((size_t)n * 3 + rIdx) * FDIM + cc] = (x > 0.0f) ? x : (__expf(x) - 1.0f);
}

__global__ void zero_wpart_kernel(float* __restrict__ wpart) {
  if (threadIdx.x < 3) wpart[threadIdx.x] = 0.0f;
}

// Semantic attention partial: WMMA 16-node x 128-col tiles of z @ sem_W1,
// then tanh(+b1)*w2 reduced and atomically accumulated per relation.
// Block = 256 threads (8 waves, one 16-col tile each). grid = (1250, 3).
__global__ void sem_attention_kernel(const float* __restrict__ z,
                                     const float* __restrict__ W1,  // [256,128]
                                     const float* __restrict__ b1,  // [128]
                                     const float* __restrict__ w2,  // [128]
                                     float* __restrict__ wpart) {
  __shared__ float sA[16 * FDIM];  // 16 KB
  __shared__ float red[256];
  const int rIdx = blockIdx.y;
  const int row0 = blockIdx.x * 16;
  const int tid  = threadIdx.x;
  const int wave = tid >> 5;
  const int lane = tid & 31;
  const int half = lane >> 4;
  const int cr   = lane & 15;

  for (int i = tid; i < 16 * FDIM; i += 256) {
    int r = i >> 8, cc = i & 255;
    int g = row0 + r;
    sA[i] = (g < N_DST) ? z[((size_t)g * 3 + rIdx) * FDIM + cc] : 0.0f;
  }
  __syncthreads();

  v8f c = {};
  c = wmma16x16_f32(sA, FDIM, W1 + wave * 16, 128, FDIM, c);

  int col = wave * 16 + cr;
  float bb = b1[col], ww = w2[col];
  float acc = 0.0f;
#pragma unroll
  for (int j = 0; j < 8; j++) {
    int m = j + 8 * half;
    if (row0 + m < N_DST) acc += tanhf(c[j] + bb) * ww;
  }
  red[tid] = acc;
  __syncthreads();
  for (int st = 128; st > 0; st >>= 1) {
    if (tid < st) red[tid] += red[tid + st];
    __syncthreads();
  }
  if (tid == 0) atomicAdd(&wpart[rIdx], red[0]);
}

__global__ void softmax3_kernel(const float* __restrict__ wpart,
                                float* __restrict__ att,
                                float* __restrict__ out_att, float invN) {
  float w0 = wpart[0] * invN, w1 = wpart[1] * invN, w2v = wpart[2] * invN;
  float mx = fmaxf(w0, fmaxf(w1, w2v));
  float e0 = __expf(w0 - mx), e1 = __expf(w1 - mx), e2 = __expf(w2v - mx);
  float s = e0 + e1 + e2;
  att[0] = e0 / s; att[1] = e1 / s; att[2] = e2 / s;
  out_att[0] = att[0]; out_att[1] = att[1]; out_att[2] = att[2];
}

__global__ void combine_z_kernel(const float* __restrict__ z,
                                 const float* __restrict__ att,
                                 float* __restrict__ out) {
  int i = blockIdx.x * blockDim.x + threadIdx.x;
  if (i >= N_DST * FDIM) return;
  int n = i >> 8, cc = i & 255;
  const float* zp = z + (size_t)n * 3 * FDIM + cc;
  out[i] = att[0] * zp[0] + att[1] * zp[FDIM] + att[2] * zp[2 * FDIM];
}

// ---------------------------------------------------------------------------
extern "C" void kernel_launch(void* const* d_in, const int* in_sizes, int n_in,
                              void* d_out, int out_size, void* d_ws, size_t ws_size,
                              hipStream_t stream) {
  const float* dst_feat = (const float*)d_in[0];
  struct Rel {
    const float *src_feat, *gat_W, *gat_al, *gat_ar, *gat_b, *Wt, *bt;
    const int *si, *di;
  } rel[3];
  for (int r = 0; r < 3; r++) {
    int base = 1 + r * 9;
    rel[r].src_feat = (const float*)d_in[base + 0];
    rel[r].si       = (const int*)  d_in[base + 1];
    rel[r].di       = (const int*)  d_in[base + 2];
    rel[r].gat_W    = (const float*)d_in[base + 3];
    rel[r].gat_al   = (const float*)d_in[base + 4];
    rel[r].gat_ar   = (const float*)d_in[base + 5];
    rel[r].gat_b    = (const float*)d_in[base + 6];
    rel[r].Wt       = (const float*)d_in[base + 7];
    rel[r].bt       = (const float*)d_in[base + 8];
  }
  const float* Wt_dst = (const float*)d_in[28];
  const float* bt_dst = (const float*)d_in[29];
  const float* sem_W1 = (const float*)d_in[30];
  const float* sem_b1 = (const float*)d_in[31];
  const float* sem_w2 = (const float*)d_in[32];

  // workspace carve-out (float units, 256B aligned slices)
  float* ws = (float*)d_ws;
  size_t off = 0;
  auto carve = [&](size_t n) {
    float* p = ws + off;
    off += (n + 63) & ~(size_t)63;
    return p;
  };
  float* zbuf   = carve((size_t)N_DST * 3 * FDIM);  // [N_DST,3,256]
  float* Fs     = carve((size_t)N_SRC * FDIM);      // per-relation reuse
  float* el     = carve((size_t)N_SRC * NH);
  float* er     = carve((size_t)N_DST * NH);
  float* ebuf   = carve((size_t)N_EDGE * NH);
  int*   mbuf   = (int*)carve((size_t)N_DST * NH);
  float* sbuf   = carve((size_t)N_DST * NH);
  float* outacc = carve((size_t)N_DST * FDIM);
  float* wpart  = carve(64);
  float* att    = carve(64);

  zero_wpart_kernel<<<1, 32, 0, stream>>>(wpart);

  for (int r = 0; r < 3; r++) {
    // src-side: h_src -> fs -> el   (stores Fs)
    node_transform_kernel<<<(N_SRC + 15) / 16, 128, 0, stream>>>(
        rel[r].src_feat, rel[r].Wt, rel[r].bt, rel[r].gat_W, rel[r].gat_al,
        Fs, el, N_SRC);
    // dst-side: h_dst -> fd -> er   (no F store)
    node_transform_kernel<<<(N_DST + 15) / 16, 128, 0, stream>>>(
        dst_feat, Wt_dst, bt_dst, rel[r].gat_W, rel[r].gat_ar,
        nullptr, er, N_DST);

    init_relation_kernel<<<(N_DST * FDIM + 255) / 256, 256, 0, stream>>>(
        mbuf, sbuf, outacc);

    edge_logits_max_kernel<<<(N_EDGE + 255) / 256, 256, 0, stream>>>(
        rel[r].si, rel[r].di, el, er, ebuf, mbuf);
    edge_softmax_sum_kernel<<<(N_EDGE + 255) / 256, 256, 0, stream>>>(
        rel[r].di, ebuf, mbuf, sbuf);
    edge_aggregate_kernel<<<((size_t)N_EDGE * 32 + 255) / 256, 256, 0, stream>>>(
        rel[r].si, rel[r].di, ebuf, sbuf, Fs, outacc);

    finalize_z_kernel<<<(N_DST * FDIM + 255) / 256, 256, 0, stream>>>(
        outacc, rel[r].gat_b, zbuf, r);
  }

  dim3 semGrid((N_DST + 15) / 16, 3);
  sem_attention_kernel<<<semGrid, 256, 0, stream>>>(zbuf, sem_W1, sem_b1,
                                                    sem_w2, wpart);

  float* out_z   = (float*)d_out;
  float* out_att = out_z + (size_t)N_DST * FDIM;
  softmax3_kernel<<<1, 1, 0, stream>>>(wpart, att, out_att, 1.0f / N_DST);
  combine_z_kernel<<<(N_DST * FDIM + 255) / 256, 256, 0, stream>>>(zbuf, att,
                                                                   out_z);
}